// QuantumLayer_65532611002583
// MI455X (gfx1250) — compile-verified
//
#include <hip/hip_runtime.h>

// Quantum statevector simulator (12 qubits, 4 layers) for MI455X / gfx1250.
// One workgroup (256 threads, 8 wave32) per batch sample; the 4096-amplitude
// complex64 state lives entirely in LDS. All single-qubit gates + CNOTs are
// fused into 16x16 complex matrices over 4-qubit groups and applied with
// V_WMMA_F32_16X16X4_F32 (fp32 matches the reference's complex64 precision).
// 12 full-state sweeps total instead of the reference's ~152.

typedef float wf2 __attribute__((ext_vector_type(2)));
typedef float wf8 __attribute__((ext_vector_type(8)));

#define NQ 12
#define NL 4

__device__ __forceinline__ float2 cmul(float2 a, float2 b) {
    return make_float2(a.x * b.x - a.y * b.y, a.x * b.y + a.y * b.x);
}

__global__ __launch_bounds__(256) void qsim_kernel(const float* __restrict__ x,
                                                   const float* __restrict__ w,
                                                   float* __restrict__ out) {
    __shared__ float2 st[4096];        // statevector, 32 KB
    __shared__ float  mR[5][16][16];   // per-layer group matrices (real)
    __shared__ float  mI[5][16][16];   // per-layer group matrices (imag)
    __shared__ float2 g2[NL][NQ][4];   // 2x2 complex gates G = RZ*RY
    __shared__ float  red[8][12];      // cross-wave reduction scratch

    const int tid  = threadIdx.x;
    const int b    = blockIdx.x;
    const int lane = tid & 31;
    const int wid  = tid >> 5;
    const int half = lane >> 4;   // 0: lanes 0-15, 1: lanes 16-31
    const int n16  = lane & 15;

    // ---- per-qubit 2x2 gates: G = RZ(phi) * RY(theta) ----
    // Layer 0 absorbs the angle encoding: RY(w0)*RY(x) = RY(w0 + x).
    if (tid < NL * NQ) {
        int l = tid / NQ, q = tid % NQ;
        float th = w[(l * NQ + q) * 2 + 0] + (l == 0 ? x[b * NQ + q] : 0.f);
        float ph = w[(l * NQ + q) * 2 + 1];
        float c = cosf(0.5f * th), s = sinf(0.5f * th);
        float er = cosf(0.5f * ph), ei = -sinf(0.5f * ph);   // e^{-i phi/2}
        g2[l][q][0] = make_float2( c * er,  c * ei);         // G00 = c*e
        g2[l][q][1] = make_float2(-s * er, -s * ei);         // G01 = -s*e
        g2[l][q][2] = make_float2( s * er, -s * ei);         // G10 = s*conj(e)
        g2[l][q][3] = make_float2( c * er, -c * ei);         // G11 = c*conj(e)
    }

    // ---- init |0...0> ----
    for (int i = tid; i < 4096; i += 256)
        st[i] = make_float2(i == 0 ? 1.f : 0.f, 0.f);

    for (int l = 0; l < NL; ++l) {
        __syncthreads();
        // ---- build 5 16x16 complex matrices: M_A, M_B(b=0/1), M_C(b=0/1) ----
        // M = P_chain * X_msb^b * kron(G_q0..G_q3);  M[i][j] = G[invperm(i)^8b][j]
        for (int e = tid; e < 5 * 256; e += 256) {
            int m = e >> 8, idx = e & 255, i = idx >> 4, j = idx & 15;
            int g    = (m == 0) ? 0 : (m <= 2 ? 1 : 2);
            int bvar = (m == 0) ? 0 : (m <= 2 ? (m - 1) : (m - 3));
            int ip = i;
            ip ^= (ip >> 1) & 1;   // inverse of intra-group CNOT-chain permutation
            ip ^= (ip >> 1) & 2;
            ip ^= (ip >> 1) & 4;
            ip ^= bvar << 3;       // conditional X from cross-group CNOT target
            float2 v = make_float2(1.f, 0.f);
            #pragma unroll
            for (int k = 0; k < 4; ++k) {
                int bi = (ip >> (3 - k)) & 1, bj = (j >> (3 - k)) & 1;
                v = cmul(v, g2[l][4 * g + k][2 * bi + bj]);
            }
            mR[m][i][j] = v.x;
            mI[m][i][j] = v.y;
        }
        __syncthreads();

        // ---- 3 passes: group A (bits 11-8), B (bits 7-4), C (bits 3-0) ----
        #pragma unroll
        for (int pass = 0; pass < 3; ++pass) {
            #pragma unroll
            for (int t2 = 0; t2 < 2; ++t2) {
                const int tt = wid * 2 + t2;   // tile 0..15, disjoint columns
                int mm, rowmul, colbase;       // state addr(j) = colbase + j*rowmul
                if (pass == 0)      { mm = 0;            rowmul = 256; colbase = (tt << 4) | n16; }
                else if (pass == 1) { mm = 1 + (tt & 1); rowmul = 16;  colbase = (tt << 8) | n16; }
                else                { mm = 3 + (tt & 1); rowmul = 1;   // even/odd cols split by control bit
                                      colbase = (((((tt >> 1) << 4) | n16) << 1) | (tt & 1)) << 4; }

                wf8 accr = {0, 0, 0, 0, 0, 0, 0, 0};
                wf8 acci = {0, 0, 0, 0, 0, 0, 0, 0};
                #pragma unroll
                for (int kb = 0; kb < 16; kb += 4) {   // K=16 as 4x WMMA 16x16x4
                    int j0 = kb + 2 * half, j1 = j0 + 1;
                    float2 s0 = st[colbase + j0 * rowmul];
                    float2 s1 = st[colbase + j1 * rowmul];
                    wf2 br  = {s0.x, s1.x};
                    wf2 bim = {s0.y, s1.y};
                    wf2 ar  = {mR[mm][n16][j0], mR[mm][n16][j1]};
                    wf2 ai  = {mI[mm][n16][j0], mI[mm][n16][j1]};
                    wf2 an  = -ai;
                    // (Mr + iMi)(Sr + iSi): Dr = Mr*Sr - Mi*Si ; Di = Mr*Si + Mi*Sr
                    accr = __builtin_amdgcn_wmma_f32_16x16x4_f32(false, ar, false, br,  (short)0, accr, false, false);
                    accr = __builtin_amdgcn_wmma_f32_16x16x4_f32(false, an, false, bim, (short)0, accr, false, false);
                    acci = __builtin_amdgcn_wmma_f32_16x16x4_f32(false, ar, false, bim, (short)0, acci, false, false);
                    acci = __builtin_amdgcn_wmma_f32_16x16x4_f32(false, ai, false, br,  (short)0, acci, false, false);
                }
                #pragma unroll
                for (int vv = 0; vv < 8; ++vv) {       // D layout: M = vv + 8*half
                    int i = vv + 8 * half;
                    st[colbase + i * rowmul] = make_float2(accr[vv], acci[vv]);
                }
            }
            __syncthreads();
        }
    }

    // ---- expectation values <Z_q> = sum |amp|^2 * (1 - 2*bit_{11-q}) ----
    float z[NQ];
    #pragma unroll
    for (int q = 0; q < NQ; ++q) z[q] = 0.f;
    for (int k = 0; k < 16; ++k) {
        int idx = (k << 8) + tid;
        float2 a = st[idx];
        float p = a.x * a.x + a.y * a.y;
        #pragma unroll
        for (int q = 0; q < NQ; ++q)
            z[q] += ((idx >> (11 - q)) & 1) ? -p : p;
    }
    #pragma unroll
    for (int q = 0; q < NQ; ++q) {
        float v = z[q];
        #pragma unroll
        for (int off = 16; off > 0; off >>= 1)
            v += __shfl_xor(v, off, 32);   // wave32 reduction
        z[q] = v;
    }
    if (lane == 0) {
        #pragma unroll
        for (int q = 0; q < NQ; ++q) red[wid][q] = z[q];
    }
    __syncthreads();
    if (tid < NQ) {
        float s = 0.f;
        #pragma unroll
        for (int ww = 0; ww < 8; ++ww) s += red[ww][tid];
        out[b * NQ + tid] = s;
    }
}

extern "C" void kernel_launch(void* const* d_in, const int* in_sizes, int n_in,
                              void* d_out, int out_size, void* d_ws, size_t ws_size,
                              hipStream_t stream) {
    const float* x = (const float*)d_in[0];   // (B, 12) float32
    const float* w = (const float*)d_in[1];   // (4, 12, 2) float32
    float* out = (float*)d_out;               // (B, 12) float32
    int B = in_sizes[0] / NQ;
    qsim_kernel<<<B, 256, 0, stream>>>(x, w, out);
}